// SmallPooling_69243462746843
// MI455X (gfx1250) — compile-verified
//
#include <hip/hip_runtime.h>
#include <hip/hip_bf16.h>
#include <stdint.h>

typedef __attribute__((ext_vector_type(2))) float v2f;
typedef __attribute__((ext_vector_type(8))) float v8f;

#define B_  16
#define S_  4096
#define D_  512
#define H_  8
#define DH_ 64

#define ROWS_    128        // rows staged per block
#define TSTRIDE  532        // LDS row stride (floats): 532%64=20 -> m*20 mod 64
                            // distinct (phase A); 2*532%64=40 -> disjoint half-
                            // wave bank sets (phase B B-frags)
#define ESTRIDE  16         // elds stride: heads 0..7 live, 8..15 zero-padded
#define NTHREADS 256

// ---------------- ws zeroing (Z + u are atomically accumulated) -------------
__global__ void zero_ws_kernel(float* __restrict__ ws, int n) {
  int i = blockIdx.x * blockDim.x + threadIdx.x;
  if (i < n) ws[i] = 0.f;
}

// ---------------- Fused kernel --------------------------------------------
// One block = 128 rows of inputs, staged ONCE from HBM.
//  Phase 0: EACH WAVE async-stages its own 16-row subtile
//           (global_load_async_to_lds_b128, per-wave ASYNCcnt) -> DMA of one
//           wave overlaps WMMA compute of sibling waves on the same SIMD.
//           Meanwhile all threads build wT = Wa^T zero-padded to 16 cols and
//           zero the elds head padding (gated by the single barrier).
//  Phase A: logits = tile @ Wa via V_WMMA_F32_16X16X4_F32 (8 waves x 16-row
//           subtile, K=512); contiguous float2 LDS frags, conflict-free.
//           e = exp(logit*temp)*mask -> elds, partial Z -> global atomics.
//  Phase B: u[b,:,:] += e^T @ tile, also WMMA: M=16 (8 heads + zero pad),
//           K=128 rows, N=512 (4 col-tiles/wave). Lanes 0-15 commit atomics.
__global__ __launch_bounds__(NTHREADS) void fused_pool_kernel(
    const float* __restrict__ inputs, const int* __restrict__ mask,
    const float* __restrict__ wA, const float* __restrict__ temp,
    float* __restrict__ Z, float* __restrict__ u)
{
  __shared__ float tile[ROWS_ * TSTRIDE];   // 266 KB
  __shared__ float wT[16 * TSTRIDE];        // 33.25 KB  (wT[n][k], n>=8 -> 0)
  __shared__ float elds[ROWS_ * ESTRIDE];   // 8 KB

  const int t = threadIdx.x;
  const int lane = t & 31;
  const int wave = t >> 5;
  const long row_base = (long)blockIdx.x * ROWS_;      // global first row
  const int  b = (int)(row_base >> 12);                // row_base / 4096
  const int  m     = lane & 15;           // A-row / B-col within fragment
  const int  khalf = (lane >> 4) << 1;    // 0 or 2
  const int  r0    = wave * 16;           // this wave's subtile rows

  // ---- Phase 0a: per-wave async stage of OWN 16x512 subtile (starts DMA) --
  {
    const float* __restrict__ gsub = inputs + (row_base + r0) * D_;
    float* __restrict__ lsub = tile + r0 * TSTRIDE;
    #pragma unroll
    for (int i = 0; i < (16 * D_ / 4) / 32; ++i) {     // 64 wave-instructions
      const int un = lane + i * 32;        // float4 unit in subtile
      const int r  = un >> 7;              // / (512/4)
      const int c  = (un & 127) << 2;      // float column
      const uint32_t lds_off = (uint32_t)(uintptr_t)(lsub + r * TSTRIDE + c);
      const uint64_t gaddr   = (uint64_t)(uintptr_t)(gsub + (long)r * D_ + c);
      asm volatile("global_load_async_to_lds_b128 %0, %1, off"
                   :: "v"(lds_off), "v"(gaddr) : "memory");
    }
  }
  // ---- Phase 0b: cooperative wT / elds-pad fill (independent of DMA) ------
  {
    #pragma unroll
    for (int i = 0; i < (16 * D_) / NTHREADS; ++i) {   // 32 per thread
      const int idx = t + i * NTHREADS;
      const int n = idx >> 9;                // 0..15
      const int k = idx & 511;
      wT[n * TSTRIDE + k] = (n < H_) ? wA[k * H_ + n] : 0.f;
    }
    #pragma unroll
    for (int i = 0; i < (ROWS_ * 8) / NTHREADS; ++i) { // heads 8..15 -> 0
      const int idx = t + i * NTHREADS;      // 0..1023
      const int r = idx >> 3;
      const int h = (idx & 7) + 8;
      elds[r * ESTRIDE + h] = 0.f;
    }
  }
  __syncthreads();                           // gates wT/elds-pad only
  asm volatile("s_wait_asynccnt 0x0" ::: "memory");    // own subtile resident

  // ---- Phase A: logits via fp32 WMMA; e -> elds; partial Z atomics ----
  {
    const float* __restrict__ arow = tile + (r0 + m) * TSTRIDE + khalf;
    const float* __restrict__ brow = wT + m * TSTRIDE + khalf;

    v8f c = {0.f, 0.f, 0.f, 0.f, 0.f, 0.f, 0.f, 0.f};
    #pragma unroll 4
    for (int kk = 0; kk < D_ / 4; ++kk) {
      const float2 av = *reinterpret_cast<const float2*>(arow + kk * 4);
      const float2 bv = *reinterpret_cast<const float2*>(brow + kk * 4);
      v2f a; a.x = av.x; a.y = av.y;
      v2f bb; bb.x = bv.x; bb.y = bv.y;
      c = __builtin_amdgcn_wmma_f32_16x16x4_f32(
          false, a, false, bb, (short)0, c, false, false);
    }
    if (m < H_) {                            // lane owns head h=m
      const int   h     = m;
      const int   mbase = (lane < 16) ? 0 : 8;
      const float th    = temp[h];
      float zsum = 0.f;
      #pragma unroll
      for (int v = 0; v < 8; ++v) {
        const int  rloc = r0 + mbase + v;
        const long row  = row_base + rloc;
        const float val = __expf(c[v] * th) * (float)mask[row];
        elds[rloc * ESTRIDE + h] = val;
        zsum += val;
      }
      atomicAdd(&Z[b * H_ + h], zsum);
    }
  }
  __syncthreads();                           // full tile + all e now valid

  // ---- Phase B: u += e^T @ tile on WMMA (M=16 heads-padded, K=128, N=512) --
  {
    #pragma unroll
    for (int ct = 0; ct < 4; ++ct) {
      const int cbase = (wave * 4 + ct) * 16;      // 16-col tile
      const float* __restrict__ acol = elds + khalf * ESTRIDE + m;         // e^T
      const float* __restrict__ bcol = tile + khalf * TSTRIDE + cbase + m; // tile
      v8f cc = {0.f, 0.f, 0.f, 0.f, 0.f, 0.f, 0.f, 0.f};
      #pragma unroll 4
      for (int k4 = 0; k4 < ROWS_ / 4; ++k4) {      // 32 k-steps of 4 rows
        const int k0 = k4 * 4;
        v2f a;  a.x  = acol[k0 * ESTRIDE];
                a.y  = acol[(k0 + 1) * ESTRIDE];
        v2f bb; bb.x = bcol[k0 * TSTRIDE];
                bb.y = bcol[(k0 + 1) * TSTRIDE];
        cc = __builtin_amdgcn_wmma_f32_16x16x4_f32(
            false, a, false, bb, (short)0, cc, false, false);
      }
      if (lane < 16) {                       // rows of C = heads 0..7 (+zeros)
        float* __restrict__ up = u + (long)(b * H_) * D_ + cbase + m;
        #pragma unroll
        for (int v = 0; v < H_; ++v)
          atomicAdd(up + (long)v * D_, cc[v]);
      }
    }
  }
}

// ---------------- Kernel 3a: flat[b, h*64+d] = (u[b,h,:]/Z) . Vw[:,h,d] -----
__global__ __launch_bounds__(64) void pooled_kernel(
    const float* __restrict__ u, const float* __restrict__ Z,
    const float* __restrict__ vw, float* __restrict__ flat)
{
  const int bh = blockIdx.x;            // 128 = B*H
  const int b = bh >> 3, h = bh & 7;
  const int d = threadIdx.x;            // 64
  const float invZ = 1.f / (Z[bh] + 1e-30f);
  const float* __restrict__ ub = u + (long)bh * D_;
  float acc = 0.f;
  for (int i = 0; i < D_; ++i)
    acc += ub[i] * vw[(long)i * (H_ * DH_) + h * DH_ + d];
  flat[(long)b * (H_ * DH_) + h * DH_ + d] = acc * invZ;
}

// ---------------- Kernel 3b: out = flat @ FF + bias -------------------------
__global__ __launch_bounds__(256) void ff_kernel(
    const float* __restrict__ flat, const float* __restrict__ ff,
    const float* __restrict__ bias, float* __restrict__ out)
{
  const int b = blockIdx.x;             // 16
  const int t = threadIdx.x;            // 256, each thread 2 outputs
  float a0 = bias[t];
  float a1 = bias[t + 256];
  const float* __restrict__ fb = flat + (long)b * 512;
  for (int j = 0; j < 512; ++j) {
    const float f = fb[j];
    a0 += f * ff[(long)j * 512 + t];
    a1 += f * ff[(long)j * 512 + t + 256];
  }
  out[(long)b * 512 + t]       = a0;
  out[(long)b * 512 + t + 256] = a1;
}

// ---------------- launch ----------------------------------------------------
extern "C" void kernel_launch(void* const* d_in, const int* in_sizes, int n_in,
                              void* d_out, int out_size, void* d_ws, size_t ws_size,
                              hipStream_t stream) {
  const float* inputs = (const float*)d_in[0];   // [16,4096,512] f32
  const int*   mask   = (const int*)d_in[1];     // [16,4096] i32
  const float* wA     = (const float*)d_in[2];   // [512,8] f32
  const float* vw     = (const float*)d_in[3];   // [512,8,64] f32
  const float* temp   = (const float*)d_in[4];   // [1,1,8] f32
  const float* ff     = (const float*)d_in[5];   // [512,512] f32
  const float* bias   = (const float*)d_in[6];   // [1,512] f32
  float* out = (float*)d_out;                    // [16,512] f32
  float* ws  = (float*)d_ws;

  // ws layout (floats): Z[128] | u[65536] | flat[8192]
  float* Z    = ws;
  float* u    = ws + 128;
  float* flat = u + (long)B_ * H_ * D_;

  const int nzero = 128 + B_ * H_ * D_;          // Z + u
  zero_ws_kernel<<<(nzero + 255) / 256, 256, 0, stream>>>(ws, nzero);
  fused_pool_kernel<<<(B_ * S_) / ROWS_, NTHREADS, 0, stream>>>(
      inputs, mask, wA, temp, Z, u);
  pooled_kernel<<<B_ * H_, 64, 0, stream>>>(u, Z, vw, flat);
  ff_kernel<<<B_, 256, 0, stream>>>(flat, ff, bias, out);
}